// ParentAttentionLayer_39548058862144
// MI455X (gfx1250) — compile-verified
//
#include <hip/hip_runtime.h>

#define NROW 65536
#define NV   200
#define HD   128
#define KD   32
#define VPAD 208
#define NSTAT 15
#define NBLK_STATS 480
#define PFS  12   // pf LDS row stride (11 features + 1 zero pad column)
#define FINF __builtin_inff()

typedef __attribute__((ext_vector_type(2))) float v2f;
typedef __attribute__((ext_vector_type(8))) float v8f;

__device__ __forceinline__ float n2n(float x) {
  if (x != x) return 0.f;
  if (x >  3.0e38f) return  3.4028235e38f;
  if (x < -3.0e38f) return -3.4028235e38f;
  return x;
}
__device__ __forceinline__ float gelu_tanh(float x) {
  return 0.5f * x * (1.f + tanhf(0.7978845608028654f * (x + 0.044715f * x * x * x)));
}

// ---------------------------------------------------------------------------
// Pass 1: fused per-column streaming statistics (one HBM pass over data).
// Thread t owns column t; grid-stride over rows; deterministic partials.
// ---------------------------------------------------------------------------
__global__ void __launch_bounds__(256) k_col_stats(const float* __restrict__ data,
                                                   const int* __restrict__ pT,
                                                   float* __restrict__ part) {
  const int T = *pT;
  const int c = threadIdx.x;
  float sC = 0.f, sC2 = 0.f, sTC = 0.f, sB = 0.f, sTB = 0.f, sCB = 0.f;
  float sT2B = 0.f, sC2B = 0.f, sTCB = 0.f, sI = 0.f, sTI = 0.f, sLag = 0.f, sRev = 0.f;
  float cmn = FINF, cmx = -FINF;
  for (int i = blockIdx.x; i < NROW; i += gridDim.x) {
    const float* __restrict__ row = data + (size_t)i * (NV * 3);
    if (c < NV && i + (int)gridDim.x < NROW)
      __builtin_prefetch(data + (size_t)(i + gridDim.x) * (NV * 3) + c * 3, 0, 1);
    float tv = row[T * 3 + 0];
    float ti = row[T * 3 + 1];
    if (c < NV) {
      float cv = row[c * 3 + 0];
      float ci = row[c * 3 + 1];
      sC += cv; sC2 += cv * cv; sTC += tv * cv;
      float both = (1.f - ti) * (1.f - ci);
      sB += both; sTB += tv * both; sCB += cv * both;
      sT2B += tv * tv * both; sC2B += cv * cv * both; sTCB += tv * cv * both;
      float iv = (ci > 0.5f) ? 1.f : 0.f;
      sI += iv; sTI += tv * iv;
      if (i < NROW - 1) sLag += data[((size_t)(i + 1) * NV + T) * 3] * cv;
      if (i > 0)        sRev += data[((size_t)(i - 1) * NV + T) * 3] * cv;
      cmn = fminf(cmn, cv); cmx = fmaxf(cmx, cv);
    }
  }
  if (c < VPAD) {
    float* p = part + ((size_t)blockIdx.x * NSTAT) * VPAD + c;
    p[0 * VPAD] = sC;   p[1 * VPAD] = sC2;  p[2 * VPAD] = sTC;
    p[3 * VPAD] = sB;   p[4 * VPAD] = sTB;  p[5 * VPAD] = sCB;
    p[6 * VPAD] = sT2B; p[7 * VPAD] = sC2B; p[8 * VPAD] = sTCB;
    p[9 * VPAD] = sI;   p[10 * VPAD] = sTI; p[11 * VPAD] = sLag;
    p[12 * VPAD] = sRev; p[13 * VPAD] = cmn; p[14 * VPAD] = cmx;
  }
}

__global__ void k_reduce_stats(const float* __restrict__ part, float* __restrict__ statsF) {
  int idx = blockIdx.x * blockDim.x + threadIdx.x;
  if (idx >= NSTAT * VPAD) return;
  int s = idx / VPAD;
  float acc;
  if (s == 13) {
    acc = FINF;
    for (int b = 0; b < NBLK_STATS; ++b) acc = fminf(acc, part[(size_t)b * NSTAT * VPAD + idx]);
  } else if (s == 14) {
    acc = -FINF;
    for (int b = 0; b < NBLK_STATS; ++b) acc = fmaxf(acc, part[(size_t)b * NSTAT * VPAD + idx]);
  } else {
    acc = 0.f;
    for (int b = 0; b < NBLK_STATS; ++b) acc += part[(size_t)b * NSTAT * VPAD + idx];
  }
  statsF[idx] = acc;
}

// ---------------------------------------------------------------------------
// Pass 2: per-column exact ranks via MSD counting sort.
// 65536-bin histogram in 256KB LDS (CDNA5: 320KB/WGP). Scattered value is
// (low16key << 16) | index16: unique, and index-in-low-bits reproduces stable
// argsort tie-breaking exactly. mode 0: write target ranks (u16).
// mode 1: accumulate sum((tr-m)(cr-m)) per column.
// ---------------------------------------------------------------------------
__global__ void __launch_bounds__(256) k_rank(const float* __restrict__ data,
                                              const int* __restrict__ pT,
                                              unsigned short* __restrict__ tr,
                                              float* __restrict__ rankdot,
                                              unsigned* __restrict__ scratchAll,
                                              int mode) {
  extern __shared__ unsigned smemU[];
  unsigned* hist = smemU;              // 65536
  unsigned* stmp = smemU + 65536;      // 256
  float*    fred = (float*)(smemU + 65536 + 256); // 256
  const int T = *pT;
  unsigned* scratch = scratchAll + (size_t)blockIdx.x * NROW;
  const int tid = threadIdx.x;
  const int col0  = (mode == 0) ? T : (int)blockIdx.x;
  const int cstep = (mode == 0) ? NV : (int)gridDim.x;
  for (int col = col0; col < NV; col += cstep) {
    __syncthreads();
    for (int b = tid; b < 65536; b += 256) hist[b] = 0u;
    __syncthreads();
    for (int i = tid; i < NROW; i += 256) {
      unsigned u = __float_as_uint(data[((size_t)i * NV + col) * 3]);
      u = (u & 0x80000000u) ? ~u : (u | 0x80000000u);
      atomicAdd(&hist[u >> 16], 1u);
    }
    __syncthreads();
    { // exclusive scan of 65536 bins: 256 serial + 1 block scan + 256 serial
      const int b0 = tid * 256;
      unsigned s = 0;
      for (int k2 = 0; k2 < 256; ++k2) s += hist[b0 + k2];
      stmp[tid] = s;
      __syncthreads();
      if (tid == 0) {
        unsigned run = 0;
        for (int t2 = 0; t2 < 256; ++t2) { unsigned v = stmp[t2]; stmp[t2] = run; run += v; }
      }
      __syncthreads();
      unsigned run = stmp[tid];
      for (int k2 = 0; k2 < 256; ++k2) { unsigned v = hist[b0 + k2]; hist[b0 + k2] = run; run += v; }
    }
    __syncthreads();
    for (int i = tid; i < NROW; i += 256) {
      unsigned u = __float_as_uint(data[((size_t)i * NV + col) * 3]);
      u = (u & 0x80000000u) ? ~u : (u | 0x80000000u);
      unsigned pos = atomicAdd(&hist[u >> 16], 1u);
      scratch[pos] = ((u & 0xFFFFu) << 16) | (unsigned)i;
    }
    __threadfence();
    __syncthreads();
    // post-scatter: hist[b] == exclusive end of bin b; start(b) == hist[b-1]
    float acc = 0.f;
    for (int b = tid; b < 65536; b += 256) {
      unsigned end = hist[b];
      unsigned start = (b == 0) ? 0u : hist[b - 1];
      for (unsigned m = start; m < end; ++m) {
        unsigned val = scratch[m];
        unsigned r = start;
        for (unsigned m2 = start; m2 < end; ++m2) r += (scratch[m2] < val) ? 1u : 0u;
        unsigned oi = val & 0xFFFFu;
        if (mode == 0) tr[oi] = (unsigned short)r;
        else acc += ((float)tr[oi] - 32767.5f) * ((float)r - 32767.5f);
      }
    }
    if (mode != 0) {
      fred[tid] = acc;
      __syncthreads();
      if (tid == 0) {
        float s = 0.f;
        for (int t2 = 0; t2 < 256; ++t2) s += fred[t2];
        rankdot[col] = s;
      }
    }
  }
}

// ---------------------------------------------------------------------------
// Pass 3: finalize features + attention/MLP head. Matmuls use the CDNA5 fp32
// matrix pipe V_WMMA_F32_16X16X4_F32 (16x16 C tiles, K stepped by 4).
// K-loop split into branchless uniform ranges so the scheduler can pipeline
// fragment loads against the WMMAs (no per-iteration divergence).
// ---------------------------------------------------------------------------
__global__ void __launch_bounds__(256) k_finalize(
    const float* __restrict__ te, const float* __restrict__ ne,
    const float* __restrict__ data,
    const float* __restrict__ Wpair, const float* __restrict__ bpair,
    const float* __restrict__ Wq, const float* __restrict__ bq,
    const float* __restrict__ Wk, const float* __restrict__ bk,
    const float* __restrict__ W1, const float* __restrict__ b1,
    const float* __restrict__ lns, const float* __restrict__ lno,
    const float* __restrict__ W2, const float* __restrict__ b2,
    const float* __restrict__ W3, const float* __restrict__ b3,
    const float* __restrict__ Wd, const int* __restrict__ pT,
    const float* __restrict__ stats, const float* __restrict__ rankdot,
    float* __restrict__ out) {
  extern __shared__ float sm[];
  float* red = sm;                 // 256
  float* sA  = sm + 256;           // 16
  float* pf  = sm + 272;           // VPAD*PFS (col 11 is zero padding)
  float* qv  = pf + VPAD * PFS;    // 32
  float* sc  = qv + 32;            // VPAD
  float* h1  = sc + VPAD;          // VPAD*129 (pad 129: conflict-free per wave)
  float* h2  = h1 + VPAD * 129;    // VPAD*65
  const int tid = threadIdx.x;
  const int T = *pT;

  // scalar reductions over target column
  float aS = 0.f, aS2 = 0.f, aMn = FINF, aMx = -FINF;
  for (int i = tid; i < NROW; i += 256) {
    float tv = data[((size_t)i * NV + T) * 3];
    aS += tv; aS2 += tv * tv; aMn = fminf(aMn, tv); aMx = fmaxf(aMx, tv);
  }
  red[tid] = aS;  __syncthreads();
  if (tid == 0) { float s = 0.f; for (int t = 0; t < 256; ++t) s += red[t]; sA[0] = s; } __syncthreads();
  red[tid] = aS2; __syncthreads();
  if (tid == 0) { float s = 0.f; for (int t = 0; t < 256; ++t) s += red[t]; sA[1] = s; } __syncthreads();
  red[tid] = aMn; __syncthreads();
  if (tid == 0) { float s = FINF; for (int t = 0; t < 256; ++t) s = fminf(s, red[t]); sA[2] = s; } __syncthreads();
  red[tid] = aMx; __syncthreads();
  if (tid == 0) {
    float s = -FINF; for (int t = 0; t < 256; ++t) s = fmaxf(s, red[t]); sA[3] = s;
    sA[4] = data[(size_t)T * 3];
    sA[5] = data[((size_t)(NROW - 1) * NV + T) * 3];
  }
  __syncthreads();
  const float stv = sA[0], stv2 = sA[1], tmn = sA[2], tmx = sA[3], tv0 = sA[4], tvN = sA[5];
  const float fN = (float)NROW;

  // pairwise features
  if (tid < VPAD) {
    if (tid < NV && tid != T) {
      int j = tid;
      float Sc = stats[0*VPAD+j], Sc2 = stats[1*VPAD+j], Stc = stats[2*VPAD+j];
      float Sb = stats[3*VPAD+j], Stb = stats[4*VPAD+j], Scb = stats[5*VPAD+j];
      float St2b = stats[6*VPAD+j], Sc2b = stats[7*VPAD+j], Stcb = stats[8*VPAD+j];
      float Si = stats[9*VPAD+j], Sti = stats[10*VPAD+j];
      float Slag = stats[11*VPAD+j], Srev = stats[12*VPAD+j];
      float cmn = stats[13*VPAD+j], cmx = stats[14*VPAD+j];
      float c0 = data[(size_t)j * 3];
      float cN = data[((size_t)(NROW - 1) * NV + j) * 3];
      float n = Sb + 1e-8f;
      float tm = Stb / n, cm2 = Scb / n;
      float cov = (Stcb - tm * Scb - cm2 * Stb + tm * cm2 * Sb) / n;
      float tsd = sqrtf(fmaxf((St2b - 2.f * tm * Stb + tm * tm * Sb) / n, 0.f) + 1e-8f);
      float csd = sqrtf(fmaxf((Sc2b - 2.f * cm2 * Scb + cm2 * cm2 * Sb) / n, 0.f) + 1e-8f);
      float corr = cov / (tsd * csd + 1e-8f);
      float c2 = corr * corr;
      float mi = -0.5f * logf(1.f - fminf(c2, 0.99f) + 1e-8f);
      float overlap = fmaxf(0.f, (fminf(tmx, cmx) - fmaxf(tmn, cmn)) /
                                 (fmaxf(tmx - tmn, cmx - cmn) + 1e-8f));
      float t_int = Sti / (Si + 1e-8f);
      float t_no = (stv - Sti) / ((fN - Si) + 1e-8f);
      float effect = fabsf(t_int - t_no);
      const float M = fN - 1.f;
      float Sa = stv - tv0, Sa2 = stv2 - tv0 * tv0;
      float Sbb = Sc - cN, Sb2 = Sc2 - cN * cN;
      float lag = n2n((Slag - Sa * Sbb / M) /
                      sqrtf((Sa2 - Sa * Sa / M) * (Sb2 - Sbb * Sbb / M)));
      Sa = stv - tvN; Sa2 = stv2 - tvN * tvN;
      Sbb = Sc - c0; Sb2 = Sc2 - c0 * c0;
      float rev = n2n((Srev - Sa * Sbb / M) /
                      sqrtf((Sa2 - Sa * Sa / M) * (Sb2 - Sbb * Sbb / M)));
      float var_t = stv2 / fN - (stv / fN) * (stv / fN);
      float cmean = Sc / fN;
      float var_c = fmaxf(Sc2 / fN - cmean * cmean, 0.f);
      float coef = corr * sqrtf(fmaxf(var_t, 0.f)) / (sqrtf(var_c) + 1e-8f);
      float Er2 = (stv2 - 2.f * coef * (Stc - cmean * stv) +
                   coef * coef * (Sc2 - fN * cmean * cmean)) / fN;
      float mr = stv / fN;
      float var_ratio = (Er2 - mr * mr) / (var_t + 1e-8f);
      float rc = n2n(rankdot[j] / 2.3456248e13f); // N(N^2-1)/12
      float sgn = (corr > 0.f) ? 1.f : ((corr < 0.f) ? -1.f : 0.f);
      pf[j*PFS+0] = n2n(corr); pf[j*PFS+1] = n2n(fabsf(corr)); pf[j*PFS+2] = n2n(c2);
      pf[j*PFS+3] = n2n(mi); pf[j*PFS+4] = n2n(overlap); pf[j*PFS+5] = n2n(effect);
      pf[j*PFS+6] = rc; pf[j*PFS+7] = sgn; pf[j*PFS+8] = lag; pf[j*PFS+9] = rev;
      pf[j*PFS+10] = n2n(var_ratio); pf[j*PFS+11] = 0.f;
    } else {
      for (int f = 0; f < PFS; ++f) pf[tid * PFS + f] = 0.f;
    }
  }
  if (tid < KD) {
    float s = bq[tid];
    for (int h = 0; h < HD; ++h) s += te[h] * Wq[h * KD + tid];
    qv[tid] = s;
  }
  __syncthreads();
  // attention scores
  if (tid < VPAD) {
    float s = 0.f;
    if (tid < NV) {
      for (int k2 = 0; k2 < KD; ++k2) {
        float kvv = bk[k2] + bpair[k2];
        for (int h = 0; h < HD; ++h) kvv += ne[tid * HD + h] * Wk[h * KD + k2];
        for (int f = 0; f < 11; ++f) kvv += pf[tid * PFS + f] * Wpair[f * KD + k2];
        s += kvv * qv[k2];
      }
      s *= 0.17677669529663687f; // 1/sqrt(32)
    }
    sc[tid] = s;
  }
  __syncthreads();

  const int wid = tid >> 5, lane = tid & 31;
  const int lh = lane >> 4, l15 = lane & 15;
  // h1 = gelu(combined @ W1 + b1) via V_WMMA_F32_16X16X4_F32
  for (int mt = wid; mt < 13; mt += 8) {
    const int mrow = mt * 16 + l15;
    const int mcl  = (mrow < NV) ? mrow : 0;     // clamped row
    const float msk = (mrow < NV) ? 1.f : 0.f;   // branchless row mask
    const float* __restrict__ nrow = ne + (size_t)mcl * HD;
    const float* __restrict__ prow = pf + mcl * PFS;
    for (int nt = 0; nt < 8; ++nt) {
      const int ncol = nt * 16 + l15;
      v8f acc = {};
      // range 1: k in [0,128) -> te (broadcast row)
      for (int k0 = 0; k0 < HD; k0 += 4) {
        const int kk = k0 + 2 * lh;
        v2f a, b;
        a.x = te[kk];             a.y = te[kk + 1];
        b.x = W1[kk * HD + ncol]; b.y = W1[(kk + 1) * HD + ncol];
        acc = __builtin_amdgcn_wmma_f32_16x16x4_f32(false, a, false, b, (short)0, acc, false, false);
      }
      // range 2: k in [128,256) -> node embeddings (masked rows)
      for (int k0 = 0; k0 < HD; k0 += 4) {
        const int kk = k0 + 2 * lh;
        v2f a, b;
        a.x = msk * nrow[kk];     a.y = msk * nrow[kk + 1];
        b.x = W1[(HD + kk) * HD + ncol]; b.y = W1[(HD + kk + 1) * HD + ncol];
        acc = __builtin_amdgcn_wmma_f32_16x16x4_f32(false, a, false, b, (short)0, acc, false, false);
      }
      // range 3: k in [256,268) -> pf*10 (LDS, zero-padded col 11)
      for (int k0 = 0; k0 < PFS; k0 += 4) {
        const int kk = k0 + 2 * lh;
        v2f a, b;
        a.x = msk * 10.f * prow[kk];
        a.y = msk * 10.f * prow[kk + 1];
        b.x = (kk     < 11) ? W1[(2 * HD + kk) * HD + ncol]     : 0.f;
        b.y = (kk + 1 < 11) ? W1[(2 * HD + kk + 1) * HD + ncol] : 0.f;
        acc = __builtin_amdgcn_wmma_f32_16x16x4_f32(false, a, false, b, (short)0, acc, false, false);
      }
      for (int v = 0; v < 8; ++v) {
        int row = mt * 16 + v + 8 * lh;
        int ch = ncol;
        h1[row * 129 + ch] = gelu_tanh(acc[v] + b1[ch]);
      }
    }
  }
  __syncthreads();
  // layernorm per row
  if (tid < VPAD) {
    float mu = 0.f;
    for (int h = 0; h < HD; ++h) mu += h1[tid * 129 + h];
    mu /= (float)HD;
    float var = 0.f;
    for (int h = 0; h < HD; ++h) { float d = h1[tid * 129 + h] - mu; var += d * d; }
    var /= (float)HD;
    float inv = rsqrtf(var + 1e-5f);
    for (int h = 0; h < HD; ++h)
      h1[tid * 129 + h] = (h1[tid * 129 + h] - mu) * inv * lns[h] + lno[h];
  }
  __syncthreads();
  // h2 = gelu(h1 @ W2 + b2)
  for (int mt = wid; mt < 13; mt += 8) {
    const int mrow = mt * 16 + l15;
    for (int nt = 0; nt < 4; ++nt) {
      const int ncol = nt * 16 + l15;
      v8f acc = {};
      for (int k0 = 0; k0 < HD; k0 += 4) {
        const int kk = k0 + 2 * lh;
        v2f a, b;
        a.x = h1[mrow * 129 + kk];
        a.y = h1[mrow * 129 + kk + 1];
        b.x = W2[kk * 64 + ncol];
        b.y = W2[(kk + 1) * 64 + ncol];
        acc = __builtin_amdgcn_wmma_f32_16x16x4_f32(false, a, false, b, (short)0, acc, false, false);
      }
      for (int v = 0; v < 8; ++v) {
        int row = mt * 16 + v + 8 * lh;
        h2[row * 65 + ncol] = gelu_tanh(acc[v] + b2[ncol]);
      }
    }
  }
  __syncthreads();
  if (tid < NV) {
    float net = b3[0];
    for (int h = 0; h < 64; ++h) net += h2[tid * 65 + h] * W3[h];
    float direct = pf[tid*PFS+0] * Wd[0] + pf[tid*PFS+8] * Wd[1] + pf[tid*PFS+10] * Wd[2];
    out[tid] = sc[tid] + net + 0.5f * direct;
  }
}

extern "C" void kernel_launch(void* const* d_in, const int* in_sizes, int n_in,
                              void* d_out, int out_size, void* d_ws, size_t ws_size,
                              hipStream_t stream) {
  (void)in_sizes; (void)n_in; (void)out_size;
  const float* te    = (const float*)d_in[0];
  const float* ne    = (const float*)d_in[1];
  const float* data  = (const float*)d_in[2];
  const float* Wpair = (const float*)d_in[3];
  const float* bpair = (const float*)d_in[4];
  const float* Wq    = (const float*)d_in[5];
  const float* bq    = (const float*)d_in[6];
  const float* Wk    = (const float*)d_in[7];
  const float* bk    = (const float*)d_in[8];
  const float* W1    = (const float*)d_in[9];
  const float* b1    = (const float*)d_in[10];
  const float* lns   = (const float*)d_in[11];
  const float* lno   = (const float*)d_in[12];
  const float* W2    = (const float*)d_in[13];
  const float* b2    = (const float*)d_in[14];
  const float* W3    = (const float*)d_in[15];
  const float* b3    = (const float*)d_in[16];
  const float* Wd    = (const float*)d_in[17];
  const int*   pT    = (const int*)d_in[18];
  float* out = (float*)d_out;
  char* ws = (char*)d_ws;

  size_t off = 0;
  auto alloc = [&](size_t bytes) { size_t o = off; off = (off + bytes + 255) & ~(size_t)255; return o; };
  size_t oPart  = alloc((size_t)NBLK_STATS * NSTAT * VPAD * 4);
  size_t oStats = alloc((size_t)NSTAT * VPAD * 4);
  size_t oRdot  = alloc((size_t)VPAD * 4);
  size_t oTr    = alloc((size_t)NROW * 2);
  size_t oScr   = off;
  size_t avail = (ws_size > oScr) ? (ws_size - oScr) : 0;
  int nsort = (int)(avail / ((size_t)NROW * 4));
  if (nsort < 1) nsort = 1;
  if (nsort > 100) nsort = 100;

  float* part = (float*)(ws + oPart);
  float* statsF = (float*)(ws + oStats);
  float* rdot = (float*)(ws + oRdot);
  unsigned short* tr = (unsigned short*)(ws + oTr);
  unsigned* scr = (unsigned*)(ws + oScr);

  k_col_stats<<<NBLK_STATS, 256, 0, stream>>>(data, pT, part);
  k_reduce_stats<<<(NSTAT * VPAD + 255) / 256, 256, 0, stream>>>(part, statsF);
  size_t smemSort = (size_t)(65536 + 256) * sizeof(unsigned) + 256 * sizeof(float);
  k_rank<<<1, 256, smemSort, stream>>>(data, pT, tr, rdot, scr, 0);
  k_rank<<<nsort, 256, smemSort, stream>>>(data, pT, tr, rdot, scr, 1);
  size_t smemFin = (size_t)(256 + 16 + VPAD * PFS + 32 + VPAD + VPAD * 129 + VPAD * 65) * sizeof(float);
  k_finalize<<<1, 256, smemFin, stream>>>(te, ne, data, Wpair, bpair, Wq, bq, Wk, bk,
                                          W1, b1, lns, lno, W2, b2, W3, b3, Wd, pT,
                                          statsF, rdot, out);
}